// GATLayer_18116172054884
// MI455X (gfx1250) — compile-verified
//
#include <hip/hip_runtime.h>

typedef __attribute__((ext_vector_type(8))) float v8f;
typedef __attribute__((ext_vector_type(2))) float v2f;

#define ALPHA 0.2f
#define NB 32
#define NN 1024
#define NF1 6
#define FD 64
#define PAD 68   // padded LDS row stride (floats): 272B, 16B-aligned, conflict-free

// workspace layout (in floats)
#define H_OFF   0
#define S1_OFF  (NB * NN * FD)           // 2,097,152
#define S2_OFF  (S1_OFF + NB * NN)       // + 32,768
#define JC_OFF  (S2_OFF + NB * NN)       // float2 per (b,j): {s2, M + ln D}
#define HT_OFF  (JC_OFF + 2 * NB * NN)   // h transposed: [b][o][j]

__device__ __forceinline__ float lrelu(float x) { return fmaxf(x, ALPHA * x); }

// CDNA5 async global->LDS copy, 16B per lane, tracked by ASYNCcnt.
__device__ __forceinline__ void async_ld_b128(uint32_t lds_byte_off, const void* gptr) {
    asm volatile("global_load_async_to_lds_b128 %0, %1, off"
                 :: "v"(lds_byte_off), "v"((unsigned long long)(size_t)gptr)
                 : "memory");
}
__device__ __forceinline__ void wait_async0() {
    asm volatile("s_wait_asynccnt 0" ::: "memory");
}
__device__ __forceinline__ uint32_t lds_off(const void* p) {
    return (uint32_t)(size_t)p;   // LDS aperture: addr[31:0] is the LDS byte offset
}

// ---------------------------------------------------------------------------
// Kernel 1: h = concat(in1,in2) @ W   (fp32 WMMA 16x16x4)
// grid (8, 32), block 256 (8 waves); each wave -> 16x64 tile of h.
// Writes h row-major (for scores) and hT=[b][o][j] (for attention B-tiles).
// ---------------------------------------------------------------------------
__global__ __launch_bounds__(256) void gat_h_gemm(const float* __restrict__ in1,
                                                  const float* __restrict__ in2,
                                                  const float* __restrict__ W,
                                                  float* __restrict__ ws) {
    __shared__ float xs[128 * PAD];   // x rows, padded stride
    __shared__ float wlT[FD * PAD];   // W transposed: wlT[o*PAD + k]
    const int b   = blockIdx.y;
    const int rb  = blockIdx.x * 128;
    const int tid = threadIdx.x;

    // async-stage 128 x-rows (straight copy, per-lane padded LDS dest)
    const uint32_t xs0 = lds_off(&xs[0]);
#pragma unroll
    for (int p = 0; p < 8; ++p) {
        int seg = tid + p * 256;          // 2048 segments of 16B
        int r   = seg >> 4;
        int f4  = (seg & 15) << 2;        // element offset within row
        int n   = rb + r;
        const float* src = (n < NF1) ? (in1 + (b * NF1 + n) * FD + f4)
                                     : (in2 + (b * (NN - NF1) + (n - NF1)) * FD + f4);
        async_ld_b128(xs0 + (uint32_t)(r * (PAD * 4) + f4 * 4), src);
    }
    // transposed W fill (one-time 16KB, plain LDS stores)
    for (int e = tid; e < FD * FD; e += 256) {
        int k = e >> 6, o = e & 63;
        wlT[o * PAD + k] = W[e];
    }
    wait_async0();
    __syncthreads();

    const int lane = tid & 31;
    const int wv   = tid >> 5;
    const int m    = lane & 15;            // M (A) / N (B,C) index
    const int kh   = (lane >> 4) << 1;     // K-half: 0 or 2
    const int nloc = wv * 16 + m;

    v8f acc[4];
#pragma unroll
    for (int t = 0; t < 4; ++t) acc[t] = 0.0f;

#pragma unroll
    for (int kb = 0; kb < FD; kb += 4) {
        v2f A = *(const v2f*)&xs[nloc * PAD + kb + kh];       // ds_load_b64
#pragma unroll
        for (int t = 0; t < 4; ++t) {
            v2f Bv = *(const v2f*)&wlT[(t * 16 + m) * PAD + kb + kh];  // ds_load_b64
            acc[t] = __builtin_amdgcn_wmma_f32_16x16x4_f32(
                false, A, false, Bv, (short)0, acc[t], false, false);
        }
    }

    float* h  = ws + H_OFF;
    float* hT = ws + HT_OFF;
    const int rowhalf = (lane >> 4) << 3;  // C/D: upper half-wave holds M=8..15
#pragma unroll
    for (int t = 0; t < 4; ++t)
#pragma unroll
        for (int r = 0; r < 8; ++r) {
            int row = rb + wv * 16 + rowhalf + r;
            int col = t * 16 + m;
            float v = acc[t][r];
            h [(b * NN + row) * FD + col] = v;
            hT[(b * FD + col) * NN + row] = v;
        }
}

// ---------------------------------------------------------------------------
// Kernel 2: s1 = h@a1, s2 = h@a2 ; one wave32 per row, shfl_xor reduction
// ---------------------------------------------------------------------------
__global__ __launch_bounds__(256) void gat_scores(const float* __restrict__ a,
                                                  float* __restrict__ ws) {
    const int tid  = threadIdx.x;
    const int lane = tid & 31;
    const int row  = blockIdx.x * 8 + (tid >> 5);   // row in [0, 32*1024)
    const float* h = ws + H_OFF;

    float x0 = h[row * FD + lane];
    float x1 = h[row * FD + 32 + lane];
    float p1 = x0 * a[lane] + x1 * a[32 + lane];
    float p2 = x0 * a[64 + lane] + x1 * a[96 + lane];
#pragma unroll
    for (int msk = 16; msk >= 1; msk >>= 1) {
        p1 += __shfl_xor(p1, msk, 32);
        p2 += __shfl_xor(p2, msk, 32);
    }
    if (lane == 0) {
        ws[S1_OFF + row] = p1;
        ws[S2_OFF + row] = p2;
    }
}

// ---------------------------------------------------------------------------
// Kernel 3: per (b,j) softmax-over-i stats.
// lrelu monotonic => M(j) = lrelu(max_i s1[i] + s2[j]);
// c[j] = M(j) + ln( sum_i exp(lrelu(s1[i]+s2[j]) - M(j)) )
// ---------------------------------------------------------------------------
__global__ __launch_bounds__(1024) void gat_jstats(float* __restrict__ ws) {
    __shared__ float s1l[NN];
    const int b = blockIdx.x;
    const int j = threadIdx.x;
    s1l[j] = ws[S1_OFF + b * NN + j];
    __syncthreads();

    float mx = -1e30f;
    for (int i = 0; i < NN; ++i) mx = fmaxf(mx, s1l[i]);

    float s2j = ws[S2_OFF + b * NN + j];
    float M   = lrelu(mx + s2j);
    float d   = 0.0f;
    for (int i = 0; i < NN; ++i) d += __expf(lrelu(s1l[i] + s2j) - M);

    float2* jc = (float2*)(ws + JC_OFF);
    jc[b * NN + j] = make_float2(s2j, M + __logf(d));
}

// ---------------------------------------------------------------------------
// Kernel 4: h'[b,i,:] = sum_j exp(lrelu(s1[i]+s2[j]) - c[j]) * h[b,j,:]
// B-tiles staged transposed+padded via async-to-LDS; attention regenerated
// on the fly (v_exp) inside the WMMA K-loop.
// ---------------------------------------------------------------------------
__global__ __launch_bounds__(256) void gat_attn_gemm(const float* __restrict__ ws,
                                                     float* __restrict__ out) {
    __shared__ float  hch[FD * PAD];  // transposed chunk: hch[o*PAD + jlocal]
    __shared__ float2 jcc[64];        // (s2, c) for chunk
    const int b    = blockIdx.y;
    const int tid  = threadIdx.x;
    const int lane = tid & 31;
    const int wv   = tid >> 5;
    const int m    = lane & 15;
    const int kh   = (lane >> 4) << 1;
    const int ibase = blockIdx.x * 128 + wv * 16;

    const float*  hT = ws + HT_OFF;
    const float2* jc = (const float2*)(ws + JC_OFF);
    const float s1v  = ws[S1_OFF + b * NN + ibase + m];

    const uint32_t h0 = lds_off(&hch[0]);
    const uint32_t j0 = lds_off(&jcc[0]);

    v8f acc[4];
#pragma unroll
    for (int t = 0; t < 4; ++t) acc[t] = 0.0f;

    for (int cb = 0; cb < NN; cb += 64) {
        // async-stage 64x64 hT chunk (rows of hT are j-contiguous) + jc chunk
#pragma unroll
        for (int p = 0; p < 4; ++p) {
            int seg = tid + p * 256;          // 1024 segments of 16B
            int o   = seg >> 4;
            int j4  = (seg & 15) << 2;        // element offset within chunk row
            async_ld_b128(h0 + (uint32_t)(o * (PAD * 4) + j4 * 4),
                          hT + (b * FD + o) * NN + cb + j4);
        }
        if (tid < 32)   // wave 0, fully active: 512B of (s2,c)
            async_ld_b128(j0 + (uint32_t)(tid * 16),
                          (const char*)(jc + b * NN + cb) + tid * 16);
        wait_async0();
        __syncthreads();

#pragma unroll
        for (int ks = 0; ks < 16; ++ks) {
            int jl = ks * 4 + kh;
            float2 p0 = jcc[jl];
            float2 p1 = jcc[jl + 1];
            v2f A;
            A.x = __expf(lrelu(s1v + p0.x) - p0.y);
            A.y = __expf(lrelu(s1v + p1.x) - p1.y);
#pragma unroll
            for (int t = 0; t < 4; ++t) {
                v2f Bv = *(const v2f*)&hch[(t * 16 + m) * PAD + jl];  // ds_load_b64
                acc[t] = __builtin_amdgcn_wmma_f32_16x16x4_f32(
                    false, A, false, Bv, (short)0, acc[t], false, false);
            }
        }
        __syncthreads();
    }

    // scatter to tuple layout: out1=(32,6,64) then out2=(32,1018,64)
    const int rowhalf = (lane >> 4) << 3;
#pragma unroll
    for (int t = 0; t < 4; ++t)
#pragma unroll
        for (int r = 0; r < 8; ++r) {
            int row = ibase + rowhalf + r;
            int col = t * 16 + m;
            float v = acc[t][r];
            if (row < NF1)
                out[(b * NF1 + row) * FD + col] = v;
            else
                out[NB * NF1 * FD + (b * (NN - NF1) + (row - NF1)) * FD + col] = v;
        }
}

// ---------------------------------------------------------------------------
extern "C" void kernel_launch(void* const* d_in, const int* in_sizes, int n_in,
                              void* d_out, int out_size, void* d_ws, size_t ws_size,
                              hipStream_t stream) {
    const float* in1 = (const float*)d_in[0];   // (32, 6, 64)
    const float* in2 = (const float*)d_in[1];   // (32, 1018, 64)
    const float* W   = (const float*)d_in[2];   // (64, 64)
    const float* a   = (const float*)d_in[3];   // (128, 1)
    float* out = (float*)d_out;
    float* ws  = (float*)d_ws;

    gat_h_gemm   <<<dim3(8, NB),       dim3(256),  0, stream>>>(in1, in2, W, ws);
    gat_scores   <<<dim3(NB * NN / 8), dim3(256),  0, stream>>>(a, ws);
    gat_jstats   <<<dim3(NB),          dim3(1024), 0, stream>>>(ws);
    gat_attn_gemm<<<dim3(8, NB),       dim3(256),  0, stream>>>(ws, out);
}